// KA_attention_orig_crossinf_20761871909466
// MI455X (gfx1250) — compile-verified
//
#include <hip/hip_runtime.h>
#include <hip/hip_bf16.h>

typedef __attribute__((ext_vector_type(2))) float v2f;
typedef __attribute__((ext_vector_type(8))) float v8f;

#define NUMV   8192           // flattened feature dim (h*p*d)
#define BATCH  16
#define KSPLIT 8
#define KCHUNK (NUMV / KSPLIT)   // 1024
#define NTILES (NUMV / 16)       // 512 feature tiles of 16 (one (h,p) each)

// ---------------------------------------------------------------- kernel 1
// silu(q) -> act  (16 x 8192)
__global__ __launch_bounds__(256)
void ka_silu_kernel(const float* __restrict__ q, float* __restrict__ act, int n)
{
    int i = blockIdx.x * blockDim.x + threadIdx.x;
    if (i < n) {
        float x = q[i];
        act[i] = x / (1.0f + __expf(-x));
    }
}

// ---------------------------------------------------------------- kernel 2
// Split-K GEMM:  ypart[ks][b][n] = sum_{m in Kchunk} act[b][m] * W[n][m]
// One wave32 per (16-wide n tile, K chunk).  WMMA f32 16x16x4.
//   A fragment: lane L -> row M = L%16 (batch), K pair = 2*(L/16)+{0,1}
//   B fragment: lane L -> col N = L%16 (feature), same K pair
//   C/D:        lane L -> N = L%16, VGPR r -> M = r + 8*(L/16)
// W is single-use streamed (no cross-wave reuse) -> non-temporal loads so the
// 256 MB stream does not evict L2-resident act/partials (L2 = 192 MB).
__global__ __launch_bounds__(128)
void ka_gemm_kernel(const float* __restrict__ act,   // [16][8192]
                    const float* __restrict__ W,     // [8192][8192] row-major (n, m)
                    float* __restrict__ ypart)       // [KSPLIT][16][8192]
{
    const int lane  = threadIdx.x & 31;
    const int wave  = threadIdx.x >> 5;
    const int ntile = blockIdx.x * 4 + wave;     // 0..511
    const int n0    = ntile * 16;
    const int k0    = blockIdx.y * KCHUNK;

    const int half = lane >> 4;                  // 0 or 1
    const int l16  = lane & 15;

    const float* aptr = act + (size_t)l16 * NUMV + k0 + 2 * half;
    const float* bptr = W   + (size_t)(n0 + l16) * NUMV + k0 + 2 * half;

    v8f acc = {0.f, 0.f, 0.f, 0.f, 0.f, 0.f, 0.f, 0.f};

#pragma unroll 16
    for (int k = 0; k < KCHUNK; k += 4) {
        v2f a = *(const v2f*)aptr;                               // RT: act is hot in L2
        v2f b = __builtin_nontemporal_load((const v2f*)bptr);    // NT: W streamed once
        acc = __builtin_amdgcn_wmma_f32_16x16x4_f32(
                  /*neg_a=*/false, a, /*neg_b=*/false, b,
                  /*c_mod=*/(short)0, acc, /*reuse_a=*/false, /*reuse_b=*/false);
        aptr += 4;
        bptr += 4;
    }

    float* yp = ypart + (size_t)blockIdx.y * BATCH * NUMV;
#pragma unroll
    for (int r = 0; r < 8; ++r) {
        int b = r + 8 * half;
        yp[(size_t)b * NUMV + n0 + l16] = acc[r];
    }
}

// ---------------------------------------------------------------- kernel 3
// Per (b, hp): sum K-split partials, add sine-spline KAN term, sigmoid,
// reduce over d=0..15  ->  sbuf[b][hp]   (hp = h*128 + p, 512 per batch)
__global__ __launch_bounds__(256)
void ka_spline_reduce_kernel(const float* __restrict__ ypart, // [KSPLIT][16][8192]
                             const float* __restrict__ q,     // [16][8192] (flat qf)
                             const float* __restrict__ coef,  // [8192][8]
                             const float* __restrict__ grid,  // [8]
                             const float* __restrict__ scale, // [8192]
                             float* __restrict__ sbuf)        // [16][512]
{
    int idx = blockIdx.x * blockDim.x + threadIdx.x;
    if (idx >= BATCH * NTILES) return;
    int b  = idx >> 9;        // / 512
    int hp = idx & 511;
    int nbase = hp * 16;

    float g[8];
#pragma unroll
    for (int f = 0; f < 8; ++f) g[f] = grid[f];

    float s = 0.0f;
#pragma unroll 4
    for (int d = 0; d < 16; ++d) {
        int n = nbase + d;
        float y = 0.0f;
#pragma unroll
        for (int ks = 0; ks < KSPLIT; ++ks)
            y += ypart[(size_t)ks * BATCH * NUMV + (size_t)b * NUMV + n];
        float x  = q[(size_t)b * NUMV + n];
        float sp = 0.0f;
#pragma unroll
        for (int f = 0; f < 8; ++f)
            sp += coef[n * 8 + f] * __sinf(g[f] * x);
        y += sp * scale[n];
        s += 1.0f / (1.0f + __expf(-y));     // sigmoid(y_k)
    }
    sbuf[idx] = s;
}

// ---------------------------------------------------------------- kernel 4
// out[b,h,p,p'] = softmax_{p'}(sbuf[b][h*128+p'])   (S_q term cancels in softmax)
__global__ __launch_bounds__(128)
void ka_softmax_bcast_kernel(const float* __restrict__ sbuf,  // [16][512]
                             float* __restrict__ out)         // [16][4][128][128]
{
    __shared__ float sh[128];
    const int bh = blockIdx.x;            // 0..63  (b*4 + h)
    const int t  = threadIdx.x;           // p'
    const int b  = bh >> 2;
    const int h  = bh & 3;

    float v = sbuf[b * 512 + h * 128 + t];

    // block max
    sh[t] = v;
    __syncthreads();
    for (int s = 64; s > 0; s >>= 1) {
        if (t < s) sh[t] = fmaxf(sh[t], sh[t + s]);
        __syncthreads();
    }
    float m = sh[0];
    __syncthreads();

    float e = __expf(v - m);
    sh[t] = e;
    __syncthreads();
    for (int s = 64; s > 0; s >>= 1) {
        if (t < s) sh[t] += sh[t + s];
        __syncthreads();
    }
    float res = e / sh[0];

    // broadcast over p: each thread writes its column value to all 128 rows
    float* op = out + (size_t)bh * 128 * 128;
#pragma unroll 4
    for (int p = 0; p < 128; ++p)
        op[(size_t)p * 128 + t] = res;    // coalesced across t
}

// ---------------------------------------------------------------- launcher
extern "C" void kernel_launch(void* const* d_in, const int* in_sizes, int n_in,
                              void* d_out, int out_size, void* d_ws, size_t ws_size,
                              hipStream_t stream)
{
    // setup_inputs order: q, k, v, grid, base_weight_q, base_weight_k,
    //                     coef_q, coef_k, scale_sp
    const float* q     = (const float*)d_in[0];
    const float* grid  = (const float*)d_in[3];
    const float* bwq   = (const float*)d_in[4];
    const float* coefq = (const float*)d_in[6];
    const float* scale = (const float*)d_in[8];
    float*       out   = (float*)d_out;

    // workspace layout
    char* ws = (char*)d_ws;
    float* act   = (float*)(ws);                                  // 16*8192 f32 = 512 KB
    float* ypart = (float*)(ws + (size_t)BATCH * NUMV * 4);       // 8*16*8192 f32 = 4 MB
    float* sbuf  = (float*)(ws + (size_t)(BATCH + KSPLIT * BATCH) * NUMV * 4); // 32 KB
    (void)ws_size; (void)in_sizes; (void)n_in; (void)out_size;

    // 1) silu(q)
    {
        int n = BATCH * NUMV;
        ka_silu_kernel<<<(n + 255) / 256, 256, 0, stream>>>(q, act, n);
    }
    // 2) split-K WMMA GEMM (only the q path survives the softmax)
    {
        dim3 g(NTILES / 4, KSPLIT);     // 128 x 8 blocks, 4 waves each
        ka_gemm_kernel<<<g, 128, 0, stream>>>(act, bwq, ypart);
    }
    // 3) partial-sum + spline + sigmoid + reduce over head_dim
    {
        int n = BATCH * NTILES;         // 8192
        ka_spline_reduce_kernel<<<(n + 255) / 256, 256, 0, stream>>>(
            ypart, q, coefq, grid, scale, sbuf);
    }
    // 4) softmax over p' and broadcast over p
    {
        ka_softmax_bcast_kernel<<<BATCH * 4, 128, 0, stream>>>(sbuf, out);
    }
}